// UserGraphNet_8400956031365
// MI455X (gfx1250) — compile-verified
//
#include <hip/hip_runtime.h>
#include <hip/hip_bf16.h>

// ---------------- problem constants (from reference) ----------------
constexpr int NB  = 512;   // batch
constexpr int NN  = 714;   // nodes
constexpr int NP  = 720;   // nodes padded to multiple of 16 (and of 4 for K-steps)
constexpr int NE  = 4096;  // edges per graph
constexpr int EDIM = 62;   // embedding dim
constexpr int FIN = 127;   // 2*62 + 3
constexpr int F   = 32;    // hidden width

typedef __attribute__((ext_vector_type(2))) float v2f;
typedef __attribute__((ext_vector_type(8))) float v8f;

#if __has_builtin(__builtin_amdgcn_global_load_async_to_lds_b32) && \
    __has_builtin(__builtin_amdgcn_s_wait_asynccnt)
#define USE_ASYNC_EDGE_STAGE 1
typedef __attribute__((address_space(1))) int g_int;
typedef __attribute__((address_space(3))) int l_int;
#else
#define USE_ASYNC_EDGE_STAGE 0
#endif

// ---------------- helpers ----------------
__device__ __forceinline__ float lrelu(float t) { return t > 0.f ? t : 0.01f * t; }

// scatter-aggregate: dst[col,:] += src[row,:]*dinv[row]*dinv[col], plus self loops.
// one wave per edge, lane = feature. everything (src/dst/dinv/edges) lives in LDS.
__device__ __forceinline__ void scatter32(const float* src, float* dst,
                                          const float* dinv,
                                          const int* e_row, const int* e_col,
                                          int lane, int wave)
{
  for (int e = wave; e < NE; e += 8) {
    int r = e_row[e], c = e_col[e];
    float nrm = dinv[r] * dinv[c];
    atomicAdd(&dst[c * F + lane], src[r * F + lane] * nrm);
  }
  for (int n = wave; n < NN; n += 8) {
    float nrm = dinv[n] * dinv[n];
    atomicAdd(&dst[n * F + lane], src[n * F + lane] * nrm);
  }
}

// out[NP,32] = hin[NP,32] @ w[32,32] via V_WMMA_F32_16X16X4_F32.
// A frag: lanes 0-15 -> M=lane, K=k..k+1 ; lanes 16-31 -> M=lane-16, K=k+2..k+3.
// B frag mirrors A (N across lanes, K split across lane halves).
// C/D: vgpr i holds rows i (lanes 0-15) and i+8 (lanes 16-31).
__device__ __forceinline__ void mm32_wmma(const float* hin, const float* w,
                                          float* out, int lane, int wave)
{
  const int m15  = lane & 15;
  const int kb   = (lane >> 4) << 1;           // 0 or 2
  const int rsel = (lane >> 4) << 3;           // 0 or 8
  for (int t = wave; t < (NP / 16) * 2; t += 8) {   // 45 mtiles x 2 ntiles
    const int mt = t >> 1, nt = t & 1;
    const int mrow = mt * 16 + m15;
    v8f acc = {};
#pragma unroll
    for (int k = 0; k < F; k += 4) {
      v2f a, b;
      a[0] = hin[mrow * F + k + kb];
      a[1] = hin[mrow * F + k + kb + 1];
      b[0] = w[(k + kb) * F + nt * 16 + m15];
      b[1] = w[(k + kb + 1) * F + nt * 16 + m15];
      acc = __builtin_amdgcn_wmma_f32_16x16x4_f32(false, a, false, b,
                                                  (short)0, acc, false, false);
    }
    const int rbase = mt * 16 + rsel;
#pragma unroll
    for (int i = 0; i < 8; ++i)
      out[(rbase + i) * F + nt * 16 + m15] = acc[i];
  }
}

// ---------------- kernel 1: zero-pad Wf1 [714,128] -> [720,128] ----------------
__global__ void pad_wf1_kernel(const float* __restrict__ Wf1, float* __restrict__ wf1p)
{
  int i = blockIdx.x * blockDim.x + threadIdx.x;
  if (i < NP * 128) {
    int k = i >> 7;
    wf1p[i] = (k < NN) ? Wf1[i] : 0.f;
  }
}

// ---------------- kernel 2: full 4-layer GCN for one graph, LDS-resident ----------------
__launch_bounds__(256, 1)
__global__ void gcn_kernel(const int* __restrict__ ids, const float* __restrict__ cont,
                           const int* __restrict__ edges, const float* __restrict__ emb,
                           const float* __restrict__ W1, const float* __restrict__ b1,
                           const float* __restrict__ W2, const float* __restrict__ b2,
                           const float* __restrict__ W3, const float* __restrict__ b3,
                           const float* __restrict__ W4, const float* __restrict__ b4,
                           float* __restrict__ h_all)
{
  __shared__ float sW[FIN * F];      // 16.3 KB (W1, later reused for W2/W3)
  __shared__ float bufA[NP * F];     // 90 KB
  __shared__ float bufB[NP * F];     // 90 KB
  __shared__ float sDinv[NP];
  __shared__ float sV[NP];
  __shared__ float sV2[NP];
  __shared__ int   sER[NE];          // 16 KB staged edge sources
  __shared__ int   sEC[NE];          // 16 KB staged edge targets

  const int bId  = blockIdx.x;
  const int tid  = threadIdx.x;
  const int lane = tid & 31;
  // wave index is the same for all 32 lanes: pin it to an SGPR so every
  // per-node address below is provably wave-uniform (scalar loads).
  const int wave = __builtin_amdgcn_readfirstlane(tid >> 5);

  const int* e_row_g = edges + (size_t)bId * 2 * NE;
  const int* e_col_g = e_row_g + NE;

  // ---- stage edge lists into LDS (they are re-read 5x below) ----
#if USE_ASYNC_EDGE_STAGE
  for (int i = tid; i < NE; i += 256) {
    __builtin_amdgcn_global_load_async_to_lds_b32(
        (g_int*)(uintptr_t)(e_row_g + i),
        (l_int*)(uint32_t)(uintptr_t)(sER + i),
        0, 0);
    __builtin_amdgcn_global_load_async_to_lds_b32(
        (g_int*)(uintptr_t)(e_col_g + i),
        (l_int*)(uint32_t)(uintptr_t)(sEC + i),
        0, 0);
  }
  __builtin_amdgcn_s_wait_asynccnt(0);
#else
  for (int i = tid; i < NE; i += 256) { sER[i] = e_row_g[i]; sEC[i] = e_col_g[i]; }
#endif

  // ---- W1 -> LDS ; zero degree accumulator ----
  for (int i = tid; i < FIN * F; i += 256) sW[i] = W1[i];
  for (int i = tid; i < NP; i += 256) sDinv[i] = 0.f;
  __syncthreads();

  // ---- degree -> rsqrt norm ----
  for (int e = tid; e < NE; e += 256) atomicAdd(&sDinv[sEC[e]], 1.0f);
  __syncthreads();
  for (int i = tid; i < NP; i += 256)
    sDinv[i] = (i < NN) ? rsqrtf(sDinv[i] + 1.0f) : 0.f;   // +1 self loop
  __syncthreads();

  // ---- layer 1: bufA = x @ W1, x gathered on the fly (wave-uniform rows) ----
  for (int n = wave; n < NP; n += 8) {
    float acc = 0.f;
    if (n < NN) {
      const int base = bId * NN + n;
      const int id0 = __builtin_amdgcn_readfirstlane(ids[base * 2 + 0]);
      const int id1 = __builtin_amdgcn_readfirstlane(ids[base * 2 + 1]);
      // prefetch the next iteration's embedding rows into cache
      if (n + 8 < NN) {
        int p0 = __builtin_amdgcn_readfirstlane(ids[(base + 8) * 2 + 0]);
        int p1 = __builtin_amdgcn_readfirstlane(ids[(base + 8) * 2 + 1]);
        __builtin_prefetch(emb + (size_t)p0 * EDIM, 0, 0);
        __builtin_prefetch(emb + (size_t)p1 * EDIM, 0, 0);
      }
      const float* er0 = emb + (size_t)id0 * EDIM;
      const float* er1 = emb + (size_t)id1 * EDIM;
      for (int k = 0; k < EDIM; ++k) acc += er0[k] * sW[k * F + lane];
      for (int k = 0; k < EDIM; ++k) acc += er1[k] * sW[(EDIM + k) * F + lane];
      const float* cr = cont + (size_t)base * 3;
      acc += cr[0] * sW[124 * F + lane];
      acc += cr[1] * sW[125 * F + lane];
      acc += cr[2] * sW[126 * F + lane];
    }
    bufA[n * F + lane] = acc;
  }
  __syncthreads();

  // ---- layer 1 aggregate + activation ----
  for (int i = tid; i < NP * F; i += 256) bufB[i] = 0.f;
  for (int i = tid; i < F * F; i += 256) sW[i] = W2[i];   // stage W2
  __syncthreads();
  scatter32(bufA, bufB, sDinv, sER, sEC, lane, wave);
  __syncthreads();
  for (int i = tid; i < NP * F; i += 256) {
    int n = i >> 5, f = i & 31;
    float t = bufB[i] + b1[f];
    bufB[i] = (n < NN) ? lrelu(t) : 0.f;
  }
  __syncthreads();

  // ---- layer 2: WMMA matmul, aggregate, lrelu(t)+t ----
  mm32_wmma(bufB, sW, bufA, lane, wave);
  __syncthreads();
  for (int i = tid; i < NP * F; i += 256) bufB[i] = 0.f;
  for (int i = tid; i < F * F; i += 256) sW[i] = W3[i];   // stage W3
  __syncthreads();
  scatter32(bufA, bufB, sDinv, sER, sEC, lane, wave);
  __syncthreads();
  for (int i = tid; i < NP * F; i += 256) {
    int n = i >> 5, f = i & 31;
    float t = bufB[i] + b2[f];
    bufB[i] = (n < NN) ? (lrelu(t) + t) : 0.f;
  }
  __syncthreads();

  // ---- layer 3 ----
  mm32_wmma(bufB, sW, bufA, lane, wave);
  __syncthreads();
  for (int i = tid; i < NP * F; i += 256) bufB[i] = 0.f;
  __syncthreads();
  scatter32(bufA, bufB, sDinv, sER, sEC, lane, wave);
  __syncthreads();
  for (int i = tid; i < NP * F; i += 256) {
    int n = i >> 5, f = i & 31;
    float t = bufB[i] + b3[f];
    bufB[i] = (n < NN) ? (lrelu(t) + t) : 0.f;
  }
  __syncthreads();

  // ---- layer 4: 32 -> 1 dot, scalar scatter, lrelu ----
  for (int n = tid; n < NP; n += 256) {
    float acc = 0.f;
    if (n < NN)
      for (int k = 0; k < F; ++k) acc += bufB[n * F + k] * W4[k];
    sV[n] = acc;
  }
  for (int i = tid; i < NP; i += 256) sV2[i] = 0.f;
  __syncthreads();
  for (int e = tid; e < NE; e += 256) {
    int r = sER[e], c = sEC[e];
    atomicAdd(&sV2[c], sV[r] * sDinv[r] * sDinv[c]);
  }
  for (int n = tid; n < NN; n += 256)
    atomicAdd(&sV2[n], sV[n] * sDinv[n] * sDinv[n]);
  __syncthreads();
  const float b4v = b4[0];
  for (int n = tid; n < NP; n += 256) {
    float t = sV2[n] + b4v;
    h_all[(size_t)bId * NP + n] = (n < NN) ? lrelu(t) : 0.f;   // pad cols = 0
  }
}

// ---------------- kernel 3: out = relu(A[512,K] @ Bw[K,128] + bias), fp32 WMMA ----------------
__launch_bounds__(256, 1)
__global__ void fc_kernel(const float* __restrict__ A, int lda,
                          const float* __restrict__ Bw,
                          const float* __restrict__ bias,
                          float* __restrict__ out, int K)
{
  const int lane = threadIdx.x & 31;
  const int wave = __builtin_amdgcn_readfirstlane(threadIdx.x >> 5); // n-tile
  const int mt   = blockIdx.x;                // 32 blocks cover 512 rows
  const int m15  = lane & 15;
  const int kb   = (lane >> 4) << 1;
  const int mrow = mt * 16 + m15;
  const int ncol = wave * 16 + m15;

  v8f acc = {};
  for (int k = 0; k < K; k += 4) {
    v2f a, b;
    a[0] = A[(size_t)mrow * lda + k + kb];
    a[1] = A[(size_t)mrow * lda + k + kb + 1];
    b[0] = Bw[(k + kb) * 128 + ncol];
    b[1] = Bw[(k + kb + 1) * 128 + ncol];
    acc = __builtin_amdgcn_wmma_f32_16x16x4_f32(false, a, false, b,
                                                (short)0, acc, false, false);
  }
  const float bv = bias[ncol];
  const int rbase = mt * 16 + ((lane >> 4) << 3);
#pragma unroll
  for (int i = 0; i < 8; ++i) {
    float v = acc[i] + bv;
    out[(size_t)(rbase + i) * 128 + ncol] = v > 0.f ? v : 0.f;
  }
}

// ---------------- launch ----------------
extern "C" void kernel_launch(void* const* d_in, const int* in_sizes, int n_in,
                              void* d_out, int out_size, void* d_ws, size_t ws_size,
                              hipStream_t stream)
{
  const int*   ids  = (const int*)d_in[0];
  const float* cont = (const float*)d_in[1];
  const int*   edges= (const int*)d_in[2];
  const float* emb  = (const float*)d_in[3];
  const float* W1   = (const float*)d_in[4];
  const float* b1   = (const float*)d_in[5];
  const float* W2   = (const float*)d_in[6];
  const float* b2   = (const float*)d_in[7];
  const float* W3   = (const float*)d_in[8];
  const float* b3   = (const float*)d_in[9];
  const float* W4   = (const float*)d_in[10];
  const float* b4   = (const float*)d_in[11];
  const float* Wf1  = (const float*)d_in[12];
  const float* bf1  = (const float*)d_in[13];
  const float* Wf2  = (const float*)d_in[14];
  const float* bf2  = (const float*)d_in[15];

  float* ws    = (float*)d_ws;
  float* h_all = ws;                          // [512,720]
  float* wf1p  = h_all + (size_t)NB * NP;     // [720,128]
  float* out1  = wf1p + (size_t)NP * 128;     // [512,128]
  float* out   = (float*)d_out;               // [512,128]

  hipLaunchKernelGGL(pad_wf1_kernel, dim3((NP * 128 + 255) / 256), dim3(256), 0, stream,
                     Wf1, wf1p);
  hipLaunchKernelGGL(gcn_kernel, dim3(NB), dim3(256), 0, stream,
                     ids, cont, edges, emb, W1, b1, W2, b2, W3, b3, W4, b4, h_all);
  hipLaunchKernelGGL(fc_kernel, dim3(32), dim3(256), 0, stream,
                     h_all, NP, wf1p, bf1, out1, NP);
  hipLaunchKernelGGL(fc_kernel, dim3(32), dim3(256), 0, stream,
                     out1, 128, Wf2, bf2, out, 128);
}